// KGAT_84293028151721
// MI455X (gfx1250) — compile-verified
//
#include <hip/hip_runtime.h>
#include <hip/hip_bf16.h>
#include <math.h>

typedef __attribute__((ext_vector_type(2))) float v2f;
typedef __attribute__((ext_vector_type(8))) float v8f;

#define NEG_SLOPE 0.01f

// ---------------------------------------------------------------------------
// 1) CSR row pointers from sorted COO rows: rp[r] = lower_bound(edge_row, r)
// ---------------------------------------------------------------------------
__global__ void build_rowptr(const int* __restrict__ erow, int* __restrict__ rp,
                             int n, int e) {
    int r = blockIdx.x * blockDim.x + threadIdx.x;
    if (r > n) return;
    int lo = 0, hi = e;
    while (lo < hi) {
        int mid = (lo + hi) >> 1;
        if (erow[mid] < r) lo = mid + 1; else hi = mid;
    }
    rp[r] = lo;
}

// ---------------------------------------------------------------------------
// 2) SpMM: y[r,:] = sum_e val[e] * x[col[e],:]   (one block per row,
//    blockDim = D so lane d reads x[col*D + d] fully coalesced)
// ---------------------------------------------------------------------------
template <int D>
__global__ void spmm_kernel(const float* __restrict__ x,
                            const float* __restrict__ vals,
                            const int* __restrict__ cols,
                            const int* __restrict__ rp,
                            float* __restrict__ y, int n) {
    int r = blockIdx.x;
    if (r >= n) return;
    int d = threadIdx.x;
    int e0 = rp[r], e1 = rp[r + 1];
    float acc = 0.f;
    for (int e = e0; e < e1; ++e) {
        int   c = cols[e];
        float v = vals[e];
        acc = fmaf(v, x[(size_t)c * D + d], acc);
    }
    y[(size_t)r * D + d] = acc;
}

// ---------------------------------------------------------------------------
// 3) Fused bi-interaction layer via V_WMMA_F32_16X16X4_F32 (wave32, one wave
//    per 16-row tile):
//      sum = leaky((ego+side) @ W1^T + b1)
//      bi  = leaky((ego*side) @ W2^T + b2)
//      ego_next = sum + bi
//    A-frag layout (16x4 f32): lanes 0-15 hold K=k0,k0+1 of row M=lane;
//    lanes 16-31 hold K=k0+2,k0+3 of row M=lane-16.
//    B-frag layout (4x16 f32): lane supplies B[k][n=lane&15] for the same
//    K pair; B[k][n] = W[n][k].
//    C/D: VGPR v holds D[M = v + 8*(lane>=16)][N = lane&15].
// ---------------------------------------------------------------------------
template <int DIN, int DOUT>
__global__ void dense_bi_layer(const float* __restrict__ ego,
                               const float* __restrict__ side,
                               const float* __restrict__ W1,
                               const float* __restrict__ b1,
                               const float* __restrict__ W2,
                               const float* __restrict__ b2,
                               float* __restrict__ ego_next, int n) {
    constexpr int NC = DOUT / 16;           // output N-chunks of 16
    __shared__ float sS[16][DIN + 1];       // ego + side
    __shared__ float sP[16][DIN + 1];       // ego * side

    const int r0   = blockIdx.x * 16;
    const int lane = threadIdx.x;           // 0..31 (wave32)
    const int half = lane >> 4;             // 0 or 1
    const int l16  = lane & 15;

    // Stage S/P tile into LDS; zero-fill tail rows so WMMA inputs are defined.
    for (int i = lane; i < 16 * DIN; i += 32) {
        int m = i / DIN, c = i % DIN;
        int r = r0 + m;
        float e = 0.f, s = 0.f;
        if (r < n) {
            e = ego [(size_t)r * DIN + c];
            s = side[(size_t)r * DIN + c];
        }
        sS[m][c] = e + s;
        sP[m][c] = e * s;
    }
    __syncthreads();

    v8f accS[NC];
    v8f accP[NC];
#pragma unroll
    for (int nc = 0; nc < NC; ++nc) { accS[nc] = (v8f)0.f; accP[nc] = (v8f)0.f; }

    for (int k0 = 0; k0 < DIN; k0 += 4) {
        const int ks = k0 + half * 2;       // this lane's K pair
        v2f aS, aP;
        aS.x = sS[l16][ks];  aS.y = sS[l16][ks + 1];
        aP.x = sP[l16][ks];  aP.y = sP[l16][ks + 1];
#pragma unroll
        for (int nc = 0; nc < NC; ++nc) {
            const int nrow = nc * 16 + l16; // output neuron index
            v2f bS, bP;
            bS.x = W1[(size_t)nrow * DIN + ks];
            bS.y = W1[(size_t)nrow * DIN + ks + 1];
            bP.x = W2[(size_t)nrow * DIN + ks];
            bP.y = W2[(size_t)nrow * DIN + ks + 1];
            accS[nc] = __builtin_amdgcn_wmma_f32_16x16x4_f32(
                false, aS, false, bS, (short)0, accS[nc], false, false);
            accP[nc] = __builtin_amdgcn_wmma_f32_16x16x4_f32(
                false, aP, false, bP, (short)0, accP[nc], false, false);
        }
    }

    // Epilogue: bias + leaky on both branches, add, scatter to ego_next.
#pragma unroll
    for (int nc = 0; nc < NC; ++nc) {
        const int ncol = nc * 16 + l16;
        const float bb1 = b1[ncol];
        const float bb2 = b2[ncol];
#pragma unroll
        for (int v = 0; v < 8; ++v) {
            const int m = v + half * 8;
            const int r = r0 + m;
            if (r < n) {
                float s = accS[nc][v] + bb1;
                s = (s > 0.f) ? s : NEG_SLOPE * s;
                float p = accP[nc][v] + bb2;
                p = (p > 0.f) ? p : NEG_SLOPE * p;
                ego_next[(size_t)r * DOUT + ncol] = s + p;
            }
        }
    }
}

// ---------------------------------------------------------------------------
// 4) Output assembly: per-row (one wave32 per row)
//    out row = [re0(64) | l2n(re1)(32) | l2n(re2)(16) |
//               im0(64) | l2n(im1)(32) | l2n(im2)(16)]   (224 cols)
// ---------------------------------------------------------------------------
__global__ void assemble_out(const float* __restrict__ ego0,
                             const float* __restrict__ ego1,
                             const float* __restrict__ ego2,
                             float* __restrict__ out, int n) {
    int r = blockIdx.x;
    if (r >= n) return;
    const int lane = threadIdx.x;           // 0..31
    const float* e0 = ego0 + (size_t)r * 128;
    const float* e1 = ego1 + (size_t)r * 64;
    const float* e2 = ego2 + (size_t)r * 32;
    float* o = out + (size_t)r * 224;

    // Layer-0 halves pass through unnormalized.
    for (int c = lane; c < 64; c += 32) {
        o[c]        = e0[c];
        o[112 + c]  = e0[64 + c];
    }

    // Layer-1: 32-wide halves, full-wave reduction each.
    float vre = e1[lane];
    float vim = e1[32 + lane];
    float sre = vre * vre, sim = vim * vim;
#pragma unroll
    for (int off = 16; off > 0; off >>= 1) {
        sre += __shfl_xor(sre, off, 32);
        sim += __shfl_xor(sim, off, 32);
    }
    o[64 + lane]  = vre / fmaxf(sqrtf(sre), 1e-12f);
    o[176 + lane] = vim / fmaxf(sqrtf(sim), 1e-12f);

    // Layer-2: lanes 0-15 hold re2, lanes 16-31 hold im2; reduce within halves.
    float v2 = e2[lane];
    float s2 = v2 * v2;
#pragma unroll
    for (int off = 8; off > 0; off >>= 1) s2 += __shfl_xor(s2, off, 32);
    float nv = v2 / fmaxf(sqrtf(s2), 1e-12f);
    int col = (lane < 16) ? (96 + lane) : (208 + (lane - 16));
    o[col] = nv;
}

// ---------------------------------------------------------------------------
// Host-side orchestration
// ---------------------------------------------------------------------------
static inline size_t align_up(size_t x, size_t a) { return (x + a - 1) & ~(a - 1); }

extern "C" void kernel_launch(void* const* d_in, const int* in_sizes, int n_in,
                              void* d_out, int out_size, void* d_ws, size_t ws_size,
                              hipStream_t stream) {
    const float* ego0  = (const float*)d_in[0];
    const float* evals = (const float*)d_in[1];
    const float* W1_0  = (const float*)d_in[2];
    const float* b1_0  = (const float*)d_in[3];
    const float* W2_0  = (const float*)d_in[4];
    const float* b2_0  = (const float*)d_in[5];
    const float* W1_1  = (const float*)d_in[6];
    const float* b1_1  = (const float*)d_in[7];
    const float* W2_1  = (const float*)d_in[8];
    const float* b2_1  = (const float*)d_in[9];
    const int*   erow  = (const int*)d_in[10];
    const int*   ecol  = (const int*)d_in[11];
    float* out = (float*)d_out;

    const int N = in_sizes[0] / 128;
    const int E = in_sizes[1];

    // Workspace carve-up (<= ~90 MB for N=100K).
    char* ws = (char*)d_ws;
    size_t off = 0;
    int*   rp    = (int*)(ws + off);   off = align_up(off + sizeof(int)   * (size_t)(N + 1), 256);
    float* side  = (float*)(ws + off); off = align_up(off + sizeof(float) * (size_t)N * 128, 256);
    float* ego1  = (float*)(ws + off); off = align_up(off + sizeof(float) * (size_t)N * 64 , 256);
    float* ego2  = (float*)(ws + off); off = align_up(off + sizeof(float) * (size_t)N * 32 , 256);
    (void)ws_size;

    const int tiles = (N + 15) / 16;

    // 1) CSR row pointers
    build_rowptr<<<(N + 1 + 255) / 256, 256, 0, stream>>>(erow, rp, N, E);

    // 2) Layer 0: side = A @ ego0 (all 128 cols at once; spmm is columnwise)
    spmm_kernel<128><<<N, 128, 0, stream>>>(ego0, evals, ecol, rp, side, N);

    // 3) Layer 0 dense bi-interaction -> ego1 (N x 64)
    dense_bi_layer<128, 64><<<tiles, 32, 0, stream>>>(
        ego0, side, W1_0, b1_0, W2_0, b2_0, ego1, N);

    // 4) Layer 1: side (reuse buffer) = A @ ego1
    spmm_kernel<64><<<N, 64, 0, stream>>>(ego1, evals, ecol, rp, side, N);

    // 5) Layer 1 dense bi-interaction -> ego2 (N x 32)
    dense_bi_layer<64, 32><<<tiles, 32, 0, stream>>>(
        ego1, side, W1_1, b1_1, W2_1, b2_1, ego2, N);

    // 6) Assemble 224-wide output with L2 norms
    assemble_out<<<N, 32, 0, stream>>>(ego0, ego1, ego2, out, N);
}